// MyNet_67577015435391
// MI455X (gfx1250) — compile-verified
//
#include <hip/hip_runtime.h>

// Sizes from the reference: WIDTH=16, KER=256, RANK=16, S=64, B=4
#define R_TOTAL 16384   // B*N rows
#define NPB     4096    // rows per batch
#define KERW    256
#define OIR     4096

typedef __attribute__((ext_vector_type(2))) float v2f;
typedef __attribute__((ext_vector_type(8))) float v8f;
typedef __attribute__((ext_vector_type(4))) unsigned int u32x4;
typedef __attribute__((ext_vector_type(8))) int i32x8;
typedef __attribute__((ext_vector_type(4))) int i32x4;

// CDNA5 fp32 WMMA: D(16x16,f32) = A(16x4,f32) * B(4x16,f32) + C
static __device__ __forceinline__ v8f wmma4(v2f a, v2f b, v8f c) {
  return __builtin_amdgcn_wmma_f32_16x16x4_f32(false, a, false, b, (short)0, c,
                                               false, false);
}

// TDM: DMA `nelem` f32 (1-D tile) from global memory into LDS.
// D# layout per CDNA5 ISA 8.3/8.4: group0 = {count, lds_addr, global_addr,
// type=2}; group1 = {data_size=4B, tensor_dim0, tensor_dim1=1, tile_dim0,
// stride0}. Groups 2/3 zero (<=2-D). Issued once (caller gates to one wave);
// completion tracked via TENSORcnt.
static __device__ __forceinline__ void tdm_load_f32(void* lds_dst,
                                                    const float* gsrc,
                                                    unsigned nelem) {
#if __has_builtin(__builtin_amdgcn_tensor_load_to_lds)
  unsigned long long ga = (unsigned long long)(const void*)gsrc;
  unsigned ldsoff = (unsigned)(unsigned long long)lds_dst;  // addr[31:0] = LDS offset
  u32x4 g0;
  g0.x = 1u;                                   // count=1, user descriptor
  g0.y = ldsoff;                               // lds_addr (bytes)
  g0.z = (unsigned)(ga & 0xffffffffull);       // global_addr[31:0]
  g0.w = (unsigned)(ga >> 32) | (2u << 30);    // global_addr[56:32], type=2
  i32x8 g1;
  g1[0] = 0x00020000;                          // data_size = 2 (4 bytes)
  g1[1] = (int)(nelem << 16);                  // tensor_dim0[15:0]
  g1[2] = (int)(nelem >> 16) | 0x00010000;     // tensor_dim0[31:16], tensor_dim1=1
  g1[3] = (int)(nelem << 16);                  // tile_dim0 (16-bit)
  g1[4] = 0;                                   // tile_dim1/2 unused
  g1[5] = (int)nelem;                          // tensor_dim0_stride[31:0]
  g1[6] = 0;
  g1[7] = 0;
  i32x4 z4 = {0, 0, 0, 0};
  i32x8 z8 = {0, 0, 0, 0, 0, 0, 0, 0};
  // 6-arg toolchain form: (g0, g1, g2, g3, pad, cpol)
  __builtin_amdgcn_tensor_load_to_lds(g0, g1, z4, z4, z8, 0);
  __builtin_amdgcn_s_wait_tensorcnt(0);
#else
  // Fallback: plain cooperative copy (single-wave caller: 32 lanes)
  unsigned t = threadIdx.x & 31;
  float* d = (float*)lds_dst;
  for (unsigned i = t; i < nelem; i += 32) d[i] = gsrc[i];
#endif
}

// ---------------- fc0: v = x @ W(12x16) + b  (scalar, trivial) -------------
__global__ void k_fc0(const float* __restrict__ x, const float* __restrict__ w,
                      const float* __restrict__ bias, float* __restrict__ v) {
  int row = blockIdx.x * blockDim.x + threadIdx.x;
  if (row >= R_TOTAL) return;
  float xi[12];
#pragma unroll
  for (int k = 0; k < 12; ++k) xi[k] = x[row * 12 + k];
#pragma unroll
  for (int c = 0; c < 16; ++c) {
    float acc = bias[c];
#pragma unroll
    for (int k = 0; k < 12; ++k) acc += xi[k] * w[k * 16 + c];
    v[row * 16 + c] = acc;
  }
}

// ------- hidden layer: h = relu(v[R,16] @ W1[16,256] + b1)  (WMMA) ---------
// 1 workgroup per 16-row m-tile; 8 waves x 2 n-tiles; W1 staged in LDS by the
// Tensor Data Mover (one tensor_load_to_lds per workgroup), b1 by plain loads.
__global__ void __launch_bounds__(256) k_hidden16x256(
    const float* __restrict__ v, const float* __restrict__ w1,
    const float* __restrict__ b1, float* __restrict__ h) {
  __shared__ float w_lds[16 * KERW];
  __shared__ float b_lds[KERW];
  int tid = threadIdx.x;
  if (tid < 32) {
    tdm_load_f32(w_lds, w1, 16 * KERW);  // wave 0 issues + waits TENSORcnt
  }
  b_lds[tid] = b1[tid];
  __syncthreads();

  int m0 = blockIdx.x * 16;
  int lane = tid & 31, wave = tid >> 5;
  int lo = lane & 15, hi = lane >> 4;

  const float* vrow = v + (size_t)(m0 + lo) * 16;
  v2f a[4];
#pragma unroll
  for (int ks = 0; ks < 4; ++ks)
    a[ks] = *(const v2f*)(vrow + ks * 4 + 2 * hi);  // A[M=lo, k=2*hi+j]

#pragma unroll
  for (int t = 0; t < 2; ++t) {
    int n = (wave * 2 + t) * 16 + lo;
    float bias = b_lds[n];
    v8f c;
#pragma unroll
    for (int r = 0; r < 8; ++r) c[r] = bias;
#pragma unroll
    for (int ks = 0; ks < 4; ++ks) {
      int k = ks * 4 + 2 * hi;
      v2f bb;
      bb.x = w_lds[k * KERW + n];
      bb.y = w_lds[(k + 1) * KERW + n];
      c = wmma4(a[ks], bb, c);
    }
#pragma unroll
    for (int r = 0; r < 8; ++r)
      h[(size_t)(m0 + r + 8 * hi) * KERW + n] = fmaxf(c[r], 0.f);
  }
}

// ---------------- zero fill ------------------------------------------------
__global__ void k_zero(float* __restrict__ p, int n) {
  int i = blockIdx.x * blockDim.x + threadIdx.x;
  if (i < n) p[i] = 0.f;
}

// ---------------- sv[b,i] = sum_n v[b,n,i] ---------------------------------
__global__ void k_sv(const float* __restrict__ v, float* __restrict__ sv) {
  int b = blockIdx.x >> 4, i = blockIdx.x & 15;
  __shared__ float red[256];
  float acc = 0.f;
  for (int n = threadIdx.x; n < NPB; n += 256)
    acc += v[((size_t)b * NPB + n) * 16 + i];
  red[threadIdx.x] = acc;
  __syncthreads();
  for (int s = 128; s > 0; s >>= 1) {
    if (threadIdx.x < s) red[threadIdx.x] += red[threadIdx.x + s];
    __syncthreads();
  }
  if (threadIdx.x == 0) sv[blockIdx.x] = red[0];
}

// ------- G[b,k,i] += h_psi[b,:,k]^T @ v[b,:,i]  (WMMA, n-chunked) ----------
// grid (16 k-tiles, 16 n-chunks, 4 batches), 1 wave each; fp32 atomics.
__global__ void __launch_bounds__(32) k_G(const float* __restrict__ hpsi,
                                          const float* __restrict__ v,
                                          float* __restrict__ G) {
  int mt = blockIdx.x, nc = blockIdx.y, b = blockIdx.z;
  int lane = threadIdx.x;
  int lo = lane & 15, hi = lane >> 4;
  int kcol = mt * 16 + lo;
  const float* hp = hpsi + (size_t)b * NPB * KERW;
  const float* vb = v + (size_t)b * NPB * 16;
  v8f c = {};
  int n0 = nc * 256;
  for (int ks = 0; ks < 64; ++ks) {
    int n = n0 + ks * 4 + 2 * hi;
    v2f a, bb;
    a.x = hp[(size_t)n * KERW + kcol];        // A = h_psi^T
    a.y = hp[(size_t)(n + 1) * KERW + kcol];
    bb.x = vb[n * 16 + lo];
    bb.y = vb[(n + 1) * 16 + lo];
    c = wmma4(a, bb, c);
  }
#pragma unroll
  for (int r = 0; r < 8; ++r)
    unsafeAtomicAdd(&G[((size_t)b * KERW + mt * 16 + r + 8 * hi) * 16 + lo],
                    c[r]);
}

// --- core[b,oir] = sum_k psi_w2[k,oir]*G[b,k,i] + psi_b2[oir]*sv[b,i] ------
__global__ void k_core(const float* __restrict__ w2,
                       const float* __restrict__ b2,
                       const float* __restrict__ G,
                       const float* __restrict__ sv,
                       float* __restrict__ core) {
  int t = blockIdx.x * blockDim.x + threadIdx.x;  // 16384
  int b = t >> 12, oir = t & 4095;
  int i = (oir >> 4) & 15;
  float acc = b2[oir] * sv[b * 16 + i];
  const float* Gb = G + (size_t)b * KERW * 16 + i;
  for (int k = 0; k < KERW; ++k) acc += w2[(size_t)k * OIR + oir] * Gb[k * 16];
  core[t] = acc;
}

// --- Wt[b,k,o] = sum_{ir} phi_w2[k,o*256+ir] * core[b,o*256+ir] ------------
__global__ void k_wt(const float* __restrict__ phi_w2,
                     const float* __restrict__ core, float* __restrict__ Wt) {
  int t = blockIdx.x * blockDim.x + threadIdx.x;  // 4*256*16 = 16384
  int b = t >> 12, rem = t & 4095;
  int k = rem >> 4, o = rem & 15;
  const float* wrow = phi_w2 + (size_t)k * OIR + o * 256;
  const float* crow = core + (size_t)b * OIR + o * 256;
  float acc = 0.f;
  for (int tt = 0; tt < 256; ++tt) acc += wrow[tt] * crow[tt];
  Wt[((size_t)b * KERW + k) * 16 + o] = acc;
}

// --- cbias[b,o] = sum_{ir} phi_b2[o*256+ir] * core[b,o*256+ir] -------------
__global__ void k_cbias(const float* __restrict__ phi_b2,
                        const float* __restrict__ core,
                        float* __restrict__ cb) {
  int t = threadIdx.x;  // 64
  int b = t >> 4, o = t & 15;
  const float* brow = phi_b2 + o * 256;
  const float* crow = core + (size_t)b * OIR + o * 256;
  float acc = 0.f;
  for (int tt = 0; tt < 256; ++tt) acc += brow[tt] * crow[tt];
  cb[t] = acc;
}

// --- y = h_phi @ Wt[b] + cbias + (v @ w_w + w_b)   (WMMA, fused) -----------
__global__ void __launch_bounds__(32) k_x1x2(
    const float* __restrict__ hphi, const float* __restrict__ Wt,
    const float* __restrict__ v, const float* __restrict__ w_w,
    const float* __restrict__ w_b, const float* __restrict__ cb,
    float* __restrict__ y) {
  int m0 = blockIdx.x * 16;
  int b = m0 >> 12;  // rows per batch = 4096, tile never straddles batches
  int lane = threadIdx.x;
  int lo = lane & 15, hi = lane >> 4;
  float bias = cb[b * 16 + lo] + w_b[lo];
  v8f c;
#pragma unroll
  for (int r = 0; r < 8; ++r) c[r] = bias;

  const float* arow = hphi + (size_t)(m0 + lo) * KERW;
  const float* Wtb = Wt + (size_t)b * KERW * 16;
  for (int ks = 0; ks < 64; ++ks) {  // K = 256: h_phi @ Wt
    int k = ks * 4 + 2 * hi;
    v2f a = *(const v2f*)(arow + k);
    v2f bb;
    bb.x = Wtb[k * 16 + lo];
    bb.y = Wtb[(k + 1) * 16 + lo];
    c = wmma4(a, bb, c);
  }
  const float* vrow = v + (size_t)(m0 + lo) * 16;
#pragma unroll
  for (int ks = 0; ks < 4; ++ks) {  // K = 16: v @ w_w
    int k = ks * 4 + 2 * hi;
    v2f a = *(const v2f*)(vrow + k);
    v2f bb;
    bb.x = w_w[k * 16 + lo];
    bb.y = w_w[(k + 1) * 16 + lo];
    c = wmma4(a, bb, c);
  }
#pragma unroll
  for (int r = 0; r < 8; ++r)
    y[(size_t)(m0 + r + 8 * hi) * 16 + lo] = c[r];
}

// ---------------- BN stats: per-column sum / sumsq over 16384 rows ---------
__global__ void k_stats(const float* __restrict__ y, float* __restrict__ stats) {
  int col = threadIdx.x & 15;
  int sub = threadIdx.x >> 4;
  int base = blockIdx.x * 256;
  float s = 0.f, s2 = 0.f;
  for (int r = 0; r < 16; ++r) {
    float val = y[(size_t)(base + sub + r * 16) * 16 + col];
    s += val;
    s2 += val * val;
  }
  unsafeAtomicAdd(&stats[col], s);
  unsafeAtomicAdd(&stats[16 + col], s2);
}

// ---------------- BN apply (+optional relu), writes next v -----------------
__global__ void k_bn(const float* __restrict__ y, const float* __restrict__ stats,
                     const float* __restrict__ g, const float* __restrict__ bta,
                     float* __restrict__ v, int do_relu) {
  int t = blockIdx.x * blockDim.x + threadIdx.x;  // 262144
  int col = t & 15;
  const float invR = 1.0f / (float)R_TOTAL;
  float mean = stats[col] * invR;
  float var = stats[16 + col] * invR - mean * mean;
  float out = (y[t] - mean) * rsqrtf(var + 1e-5f) * g[col] + bta[col];
  if (do_relu) out = fmaxf(out, 0.f);
  v[t] = out;
}

// ---------------- fc1: h = relu(v[R,16] @ W[16,128] + b)  (WMMA) -----------
__global__ void __launch_bounds__(256) k_fc1(const float* __restrict__ v,
                                             const float* __restrict__ w,
                                             const float* __restrict__ bias,
                                             float* __restrict__ h) {
  __shared__ float w_lds[16 * 128];
  __shared__ float b_lds[128];
  int tid = threadIdx.x;
  if (tid < 32) {
    tdm_load_f32(w_lds, w, 16 * 128);  // TDM stage of fc1 weights
  }
  if (tid < 128) b_lds[tid] = bias[tid];
  __syncthreads();

  int m0 = blockIdx.x * 16;
  int lane = tid & 31, wave = tid >> 5;
  int lo = lane & 15, hi = lane >> 4;
  int n = wave * 16 + lo;
  const float* vrow = v + (size_t)(m0 + lo) * 16;
  float bias0 = b_lds[n];
  v8f c;
#pragma unroll
  for (int r = 0; r < 8; ++r) c[r] = bias0;
#pragma unroll
  for (int ks = 0; ks < 4; ++ks) {
    int k = ks * 4 + 2 * hi;
    v2f a = *(const v2f*)(vrow + k);
    v2f bb;
    bb.x = w_lds[k * 128 + n];
    bb.y = w_lds[(k + 1) * 128 + n];
    c = wmma4(a, bb, c);
  }
#pragma unroll
  for (int r = 0; r < 8; ++r)
    h[(size_t)(m0 + r + 8 * hi) * 128 + n] = fmaxf(c[r], 0.f);
}

// ---------------- fc2: out = h[R,128] @ w[128,1] + b -----------------------
__global__ void k_fc2(const float* __restrict__ h, const float* __restrict__ w,
                      const float* __restrict__ b, float* __restrict__ out) {
  int row = blockIdx.x * blockDim.x + threadIdx.x;
  if (row >= R_TOTAL) return;
  float acc = b[0];
  const float* hr = h + (size_t)row * 128;
  for (int k = 0; k < 128; ++k) acc += hr[k] * w[k];
  out[row] = acc;
}

// ---------------------------------------------------------------------------
// Input order (setup_inputs insertion order, leaves flattened in-order):
// 0:x 1:fc0_w 2:fc0_b, then 4 blocks of
// [phi_w1,phi_b1,phi_w2,phi_b2,psi_w1,psi_b1,psi_w2,psi_b2,w_w,w_b,bn_g,bn_b],
// then 51:fc1_w 52:fc1_b 53:fc2_w 54:fc2_b.
extern "C" void kernel_launch(void* const* d_in, const int* in_sizes, int n_in,
                              void* d_out, int out_size, void* d_ws,
                              size_t ws_size, hipStream_t stream) {
  (void)in_sizes; (void)n_in; (void)out_size; (void)ws_size;
  const float* X = (const float*)d_in[0];
  const float* fc0_w = (const float*)d_in[1];
  const float* fc0_b = (const float*)d_in[2];
  const float* fc1_w = (const float*)d_in[51];
  const float* fc1_b = (const float*)d_in[52];
  const float* fc2_w = (const float*)d_in[53];
  const float* fc2_b = (const float*)d_in[54];

  float* ws = (float*)d_ws;
  float* v = ws;                    // 262144 f
  float* y = ws + 262144;           // 262144 f
  float* big = ws + 524288;         // 4194304 f (h_psi -> h_phi -> fc1 out)
  float* G = ws + 4718592;          // 16384 f
  float* sv = ws + 4734976;         // 64 f
  float* core = ws + 4735040;       // 16384 f
  float* Wt = ws + 4751424;         // 16384 f
  float* cb = ws + 4767808;         // 64 f
  float* stats = ws + 4767872;      // 32 f   (total ~19.1 MB)

  k_fc0<<<64, 256, 0, stream>>>(X, fc0_w, fc0_b, v);

  for (int i = 0; i < 4; ++i) {
    const int base = 3 + i * 12;
    const float* phi_w1 = (const float*)d_in[base + 0];
    const float* phi_b1 = (const float*)d_in[base + 1];
    const float* phi_w2 = (const float*)d_in[base + 2];
    const float* phi_b2 = (const float*)d_in[base + 3];
    const float* psi_w1 = (const float*)d_in[base + 4];
    const float* psi_b1 = (const float*)d_in[base + 5];
    const float* psi_w2 = (const float*)d_in[base + 6];
    const float* psi_b2 = (const float*)d_in[base + 7];
    const float* w_w = (const float*)d_in[base + 8];
    const float* w_b = (const float*)d_in[base + 9];
    const float* bn_g = (const float*)d_in[base + 10];
    const float* bn_b = (const float*)d_in[base + 11];

    // psi branch: h_psi -> G -> core
    k_hidden16x256<<<1024, 256, 0, stream>>>(v, psi_w1, psi_b1, big);
    k_zero<<<64, 256, 0, stream>>>(G, 16384);
    k_zero<<<1, 32, 0, stream>>>(stats, 32);
    k_sv<<<64, 256, 0, stream>>>(v, sv);
    k_G<<<dim3(16, 16, 4), 32, 0, stream>>>(big, v, G);
    k_core<<<64, 256, 0, stream>>>(psi_w2, psi_b2, G, sv, core);
    // phi branch: h_phi (reuses `big`) -> Wt/cbias -> fused x1+x2
    k_hidden16x256<<<1024, 256, 0, stream>>>(v, phi_w1, phi_b1, big);
    k_wt<<<64, 256, 0, stream>>>(phi_w2, core, Wt);
    k_cbias<<<1, 64, 0, stream>>>(phi_b2, core, cb);
    k_x1x2<<<1024, 32, 0, stream>>>(big, Wt, v, w_w, w_b, cb, y);
    // batchnorm (+relu except last block)
    k_stats<<<64, 256, 0, stream>>>(y, stats);
    k_bn<<<1024, 256, 0, stream>>>(y, stats, bn_g, bn_b, v, (i < 3) ? 1 : 0);
  }

  k_fc1<<<1024, 256, 0, stream>>>(v, fc1_w, fc1_b, big);
  k_fc2<<<64, 256, 0, stream>>>(big, fc2_w, fc2_b, (float*)d_out);
}